// DE3_72679436583685
// MI455X (gfx1250) — compile-verified
//
#include <hip/hip_runtime.h>
#include <hip/hip_bf16.h>
#include <stdint.h>

#define NUM_BINS 256
#define TPB      256            // 8 wave32s per block
#define WAVES    (TPB / 32)
#define NBLK     1024           // 1024*256 threads -> 64 float4 iters each on 16*2048*2048
#define DEPTH    4              // async pipeline depth (ring buffers per wave)

// ---------------------------------------------------------------------------
// CDNA5 async memory->LDS copy (gfx1250). Per-lane: 16 bytes from global addr
// (64-bit VGPR pair) into LDS at the 32-bit workgroup-relative byte offset in
// the "vdst" VGPR. Tracked by ASYNCcnt; completes in order per wave.
// Non-temporal: data is streamed once, keep it out of L2's way.
// ---------------------------------------------------------------------------
__device__ __forceinline__ void async_copy_b128_to_lds(unsigned lds_off, const void* gptr) {
    unsigned long long ga = (unsigned long long)(uintptr_t)gptr;
    asm volatile("global_load_async_to_lds_b128 %0, %1, off th:TH_LOAD_NT"
                 :: "v"(lds_off), "v"(ga)
                 : "memory");
}
__device__ __forceinline__ void wait_async3() { asm volatile("s_wait_asynccnt 3" ::: "memory"); }
__device__ __forceinline__ void wait_async2() { asm volatile("s_wait_asynccnt 2" ::: "memory"); }
__device__ __forceinline__ void wait_async1() { asm volatile("s_wait_asynccnt 1" ::: "memory"); }
__device__ __forceinline__ void wait_async0() { asm volatile("s_wait_asynccnt 0" ::: "memory"); }

// Low 32 bits of a generic pointer to __shared__ data == LDS byte offset.
__device__ __forceinline__ unsigned lds_offset_of(const void* p) {
    return (unsigned)(uintptr_t)p;
}

__global__ void DE3_zero_bins_kernel(unsigned* __restrict__ g_hist) {
    g_hist[threadIdx.x] = 0u;
}

__global__ void __launch_bounds__(TPB)
DE3_hist_kernel(const float* __restrict__ img, long long n, unsigned* __restrict__ g_hist) {
    // Per-wave privatized histograms: only intra-wave collisions serialize.
    __shared__ unsigned hist[WAVES][NUM_BINS];            // 8 KB
    // Per-wave 4-deep staging ring: 32 lanes x 16B x DEPTH buffers per wave.
    __shared__ float4 stage[WAVES][DEPTH][32];            // 16 KB

    const int tid  = threadIdx.x;
    const int wave = tid >> 5;
    const int lane = tid & 31;

    for (int i = tid; i < WAVES * NUM_BINS; i += TPB)
        ((unsigned*)hist)[i] = 0u;
    __syncthreads();

    unsigned* myhist = hist[wave];

#define BIN1(xv)                                                    \
    do {                                                            \
        float _x = (xv);                                            \
        if (_x >= 0.0f && _x < 256.0f) {                            \
            atomicAdd(&myhist[(int)_x], 1u); /* ds_add_u32 */       \
        }                                                           \
    } while (0)
#define BIN4(v) do { BIN1((v).x); BIN1((v).y); BIN1((v).z); BIN1((v).w); } while (0)

    const long long n4       = n >> 2;
    const long long gthreads = (long long)gridDim.x * TPB;
    const long long gtid     = (long long)blockIdx.x * TPB + tid;
    const long long iters    = n4 / gthreads;
    const float4*   img4     = (const float4*)img;
    const float4*   p0       = img4 + gtid;               // this thread's stream base

    // Ring buffer slot b lives at lds_base + b*512 bytes (32 lanes * 16B).
    const unsigned lds_base  = lds_offset_of(&stage[wave][0][lane]);

    if (iters >= DEPTH - 1) {
        // Prologue: put DEPTH-1 copies in flight.
        #pragma unroll
        for (int j = 0; j < DEPTH - 1; ++j)
            async_copy_b128_to_lds(lds_base + (unsigned)j * 512u, p0 + (long long)j * gthreads);

        // Steady state: 4 in flight; wait<=3 lands the oldest; consume it.
        long long it = 0;
        for (; it + (DEPTH - 1) < iters; ++it) {
            async_copy_b128_to_lds(lds_base + ((unsigned)(it + DEPTH - 1) & 3u) * 512u,
                                   p0 + (it + DEPTH - 1) * gthreads);
            wait_async3();
            float4 v = stage[wave][it & 3][lane];   // ds_load_b128
            BIN4(v);
        }
        // Drain the last 3 outstanding copies.
        wait_async2(); { float4 v = stage[wave][(it + 0) & 3][lane]; BIN4(v); }
        wait_async1(); { float4 v = stage[wave][(it + 1) & 3][lane]; BIN4(v); }
        wait_async0(); { float4 v = stage[wave][(it + 2) & 3][lane]; BIN4(v); }
    } else {
        // Tiny-problem fallback: direct loads.
        for (long long it = 0; it < iters; ++it) {
            float4 v = p0[it * gthreads];
            BIN4(v);
        }
    }

    // Remainder float4s (none for 16*2048*2048 with this grid, kept general).
    const long long rembase = iters * gthreads;
    if (rembase + gtid < n4) {
        float4 v = img4[rembase + gtid];
        BIN4(v);
    }
    // Scalar tail (n % 4).
    const long long ntail = n & 3;
    if (blockIdx.x == 0 && (long long)tid < ntail) {
        BIN1(img[(n4 << 2) + tid]);
    }
#undef BIN4
#undef BIN1

    __syncthreads();
    // Merge the 8 per-wave histograms, one global atomic per bin per block.
    for (int b = tid; b < NUM_BINS; b += TPB) {
        unsigned s = 0;
        #pragma unroll
        for (int w = 0; w < WAVES; ++w) s += hist[w][b];
        if (s) atomicAdd(&g_hist[b], s);
    }
}

__global__ void DE3_entropy_kernel(const unsigned* __restrict__ g_hist,
                                   float* __restrict__ out,
                                   float inv_temp, float Bf) {
    __shared__ float sdata[NUM_BINS];
    const int t = threadIdx.x;
    const float p = (float)g_hist[t] * inv_temp;
    // term = p*log2(p) for p>0 else 0; res = -sum(term)
    sdata[t] = (p > 0.0f) ? p * log2f(p) : 0.0f;
    __syncthreads();
    #pragma unroll
    for (int s = NUM_BINS / 2; s > 0; s >>= 1) {
        if (t < s) sdata[t] += sdata[t + s];
        __syncthreads();
    }
    if (t == 0) {
        const float sum_plogp = sdata[0];       // == -res
        out[0] = Bf * (8.0f + sum_plogp);       // B * (8 - res)
    }
}

extern "C" void kernel_launch(void* const* d_in, const int* in_sizes, int n_in,
                              void* d_out, int out_size, void* d_ws, size_t ws_size,
                              hipStream_t stream) {
    const float* img = (const float*)d_in[0];
    const long long n = (long long)in_sizes[0];     // 16*2048*2048
    unsigned* g_hist = (unsigned*)d_ws;             // 256 bins of u32 scratch
    float* out = (float*)d_out;

    const long long B  = 16;                        // reference batch dim
    const float     HW = (float)(n / B);            // temp = H*W

    DE3_zero_bins_kernel<<<1, NUM_BINS, 0, stream>>>(g_hist);
    DE3_hist_kernel<<<NBLK, TPB, 0, stream>>>(img, n, g_hist);
    DE3_entropy_kernel<<<1, NUM_BINS, 0, stream>>>(g_hist, out, 1.0f / HW, (float)B);
}